// Net_time_74096775791264
// MI455X (gfx1250) — compile-verified
//
#include <hip/hip_runtime.h>
#include <hip/hip_bf16.h>

typedef float v2f __attribute__((ext_vector_type(2)));
typedef float v8f __attribute__((ext_vector_type(8)));

#define BB   16
#define TT   1024
#define VV   25
#define CIN  64
#define CO   128
#define KP   3
#define KER  9
#define NROWS (BB*TT*VV)     // 409600
#define EPSV 1e-5f

// ---- workspace float offsets ----
#define SUM_H 0
#define SQ_H  128
#define SUM_R 256
#define SQ_R  384
#define SUM_T 512
#define SQ_T  640
#define SC1   768
#define SH1   896
#define SC2   1024
#define SH2   1152
#define SC3   1280
#define SH3   1408
#define STATS_N   1536
#define HBIAS_OFF 2048                 // 25*128 = 3200
#define WSPT_OFF  6144                 // 3*64*128 = 24576   wspT[k][i][o]
#define WRT_OFF   30720                // 64*128  = 8192     wrt[i][o]
#define WTT_OFF   38912                // 1152*128 = 147456  wtt[r*128+c][o]
#define HRAW_OFF  186368               // 52428800           h_raw[b,t,v,o]

__device__ __forceinline__ v8f wmma4(v2f a, v2f b, v8f c) {
  // V_WMMA_F32_16X16X4_F32: D = A(16x4) * B(4x16) + C(16x16)
  return __builtin_amdgcn_wmma_f32_16x16x4_f32(false, a, false, b, (short)0, c,
                                               false, false);
}

// ---- Tensor Data Mover path (6-arg clang-23 builtin), manual-copy fallback ----
#if defined(__has_builtin)
#if __has_builtin(__builtin_amdgcn_tensor_load_to_lds) && \
    __has_builtin(__builtin_amdgcn_s_wait_tensorcnt)
#define USE_TDM 1
#endif
#endif

#ifdef USE_TDM
typedef unsigned int u32x4 __attribute__((ext_vector_type(4)));
typedef int i32x4v __attribute__((ext_vector_type(4)));
typedef int i32x8v __attribute__((ext_vector_type(8)));

// DMA a [rows x 128] float tile (row stride 128 floats) from global to LDS.
__device__ __forceinline__ void tdm_load_rows(const float* gsrc, void* ldsdst,
                                              int rows) {
  const unsigned long long ga = (unsigned long long)(const void*)gsrc;
  const unsigned lds = (unsigned)(unsigned long long)ldsdst;
  u32x4 g0;
  g0[0] = 1u;                                            // count=1, no gather
  g0[1] = lds;                                           // lds_addr
  g0[2] = (unsigned)(ga & 0xffffffffull);                // global_addr lo
  g0[3] = (unsigned)((ga >> 32) & 0x01ffffffull)         // global_addr hi
          | 0x80000000u;                                 // type=2 ("image")
  i32x8v g1;
  g1[0] = 0x00020000;                                    // data_size = 4B
  g1[1] = (CO << 16);                                    // tensor_dim0 = 128
  g1[2] = (rows << 16);                                  // tensor_dim1 = rows
  g1[3] = (CO << 16);                                    // tile_dim0 = 128
  g1[4] = rows;                                          // tile_dim1 = rows
  g1[5] = CO;                                            // dim0_stride = 128
  g1[6] = 0;
  g1[7] = 0;
  const i32x4v z4 = {0, 0, 0, 0};                        // 2D: groups 2/3 zero
  const i32x8v z8 = {0, 0, 0, 0, 0, 0, 0, 0};
  __builtin_amdgcn_tensor_load_to_lds(g0, g1, z4, z4, z8, 0);
}
#endif

// ============================ K0: prep ============================
__global__ void k0_prep(const float* __restrict__ adj, const float* __restrict__ edge,
                        const float* __restrict__ w_sp, const float* __restrict__ b_sp,
                        const float* __restrict__ w_tcn, const float* __restrict__ w_res,
                        float* __restrict__ ws) {
  const int tid = blockIdx.x * blockDim.x + threadIdx.x;
  const int nth = gridDim.x * blockDim.x;
  for (int i = tid; i < STATS_N; i += nth) ws[i] = 0.f;
  // h_bias[w,o] = sum_k b_sp[k,o] * sum_v A[k,v,w]
  for (int i = tid; i < VV * CO; i += nth) {
    const int w = i / CO, o = i % CO;
    float s = 0.f;
    for (int k = 0; k < KP; ++k) {
      float cs = 0.f;
      for (int v = 0; v < VV; ++v) {
        const int a = (k * VV + v) * VV + w;
        cs += adj[a] * edge[a];
      }
      s += b_sp[k * CO + o] * cs;
    }
    ws[HBIAS_OFF + i] = s;
  }
  // wspT[k][i][o] = w_sp[k][o][i]
  for (int i = tid; i < KP * CIN * CO; i += nth) {
    const int k = i / (CIN * CO), r = i % (CIN * CO);
    const int ci = r / CO, o = r % CO;
    ws[WSPT_OFF + i] = w_sp[(k * CO + o) * CIN + ci];
  }
  // wrt[i][o] = w_res[o][i]
  for (int i = tid; i < CIN * CO; i += nth) {
    const int ci = i / CO, o = i % CO;
    ws[WRT_OFF + i] = w_res[o * CIN + ci];
  }
  // wtt[j][o] = w_tcn[o][c][r], j = r*128 + c
  for (int i = tid; i < KER * CO * CO; i += nth) {
    const int j = i / CO, o = i % CO;
    const int r = j / CO, c = j % CO;
    ws[WTT_OFF + i] = w_tcn[(o * CO + c) * KER + r];
  }
}

// ================= K1: spatial graph conv + res stats =================
// block = 128 threads (4 waves), each wave handles one t; grid = B*T/4
__global__ void __launch_bounds__(128) k1_spatial(
    const float* __restrict__ x, const float* __restrict__ adj,
    const float* __restrict__ edge, const float* __restrict__ b_res,
    float* __restrict__ ws) {
  __shared__ float s_x[4][32 * CIN];        // 32KB, zero-padded rows 25..31
  __shared__ float s_acomb[KP * VV * VV];   // 7.5KB  [k][v][w]
  __shared__ float s_ytmp[4][32 * 16];      // 8KB   per-wave y bounce tile
  __shared__ float s_stats[4 * 128];        // 2KB   sumH,sqH,sumR,sqR

  const int tid  = threadIdx.x;
  const int wave = tid >> 5;
  const int lane = tid & 31;
  const int l16  = lane & 15;
  const int hi   = lane >> 4;

  const int wg   = blockIdx.x;
  const int b    = wg >> 8;            // 256 groups of 4 t per b
  const int tgrp = wg & 255;
  const int t    = tgrp * 4 + wave;

  for (int i = tid; i < KP * VV * VV; i += 128) s_acomb[i] = adj[i] * edge[i];
  for (int i = tid; i < 4 * 128; i += 128) s_stats[i] = 0.f;
  {
    const float* xp = x + ((size_t)(b * TT + t)) * VV * CIN;
    for (int e = lane; e < 32 * CIN; e += 32) {
      const int v = e >> 6;
      s_x[wave][e] = (v < VV) ? xp[e] : 0.f;
    }
  }
  __syncthreads();

  const float* wspT = ws + WSPT_OFF;
  const float* wrt  = ws + WRT_OFF;
  const float* hb   = ws + HBIAS_OFF;
  float* hraw = ws + HRAW_OFF + ((size_t)(b * TT + t)) * VV * CO;

  for (int nt = 0; nt < 8; ++nt) {
    const int c = nt * 16 + l16;
    v8f hc0 = {}; v8f hc1 = {};
    for (int k = 0; k < KP; ++k) {
      // stage 1: y(32x16) = x(32x64) @ wspT_k(64x16)
      v8f y0 = {}; v8f y1 = {};
      for (int ks = 0; ks < 16; ++ks) {
        const int i0 = ks * 4 + 2 * hi;
        v2f a0 = *(const v2f*)&s_x[wave][l16 * CIN + i0];
        v2f a1 = *(const v2f*)&s_x[wave][(l16 + 16) * CIN + i0];
        const float* bp = wspT + ((size_t)k * CIN + i0) * CO + c;
        v2f bb; bb.x = bp[0]; bb.y = bp[CO];
        y0 = wmma4(a0, bb, y0);
        y1 = wmma4(a1, bb, y1);
      }
      // D-layout -> B-layout bounce through LDS (wave-private, DS in-order)
#pragma unroll
      for (int r = 0; r < 8; ++r) {
        s_ytmp[wave][(r + 8 * hi) * 16 + l16]        = y0[r];
        s_ytmp[wave][(16 + r + 8 * hi) * 16 + l16]   = y1[r];
      }
      // stage 2: h(32x16) += A_k^T(32x28) @ y(28x16)   (K padded 25->28)
      for (int ks = 0; ks < 7; ++ks) {
        const int v0 = ks * 4 + 2 * hi;
        const int w0 = l16, w1 = l16 + 16;
        v2f a0, a1;
        a0.x = (v0     < VV) ? s_acomb[(k * VV + v0)     * VV + w0] : 0.f;
        a0.y = (v0 + 1 < VV) ? s_acomb[(k * VV + v0 + 1) * VV + w0] : 0.f;
        a1.x = (w1 < VV && v0     < VV) ? s_acomb[(k * VV + v0)     * VV + w1] : 0.f;
        a1.y = (w1 < VV && v0 + 1 < VV) ? s_acomb[(k * VV + v0 + 1) * VV + w1] : 0.f;
        v2f bb;
        bb.x = s_ytmp[wave][v0 * 16 + l16];
        bb.y = s_ytmp[wave][(v0 + 1) * 16 + l16];
        hc0 = wmma4(a0, bb, hc0);
        hc1 = wmma4(a1, bb, hc1);
      }
    }
    // finalize h tile: + h_bias, stats, store to h_raw
    {
      float ls = 0.f, lq = 0.f;
#pragma unroll
      for (int r = 0; r < 8; ++r) {
        const int w = r + 8 * hi;                    // 0..15, always valid
        const float hv = hc0[r] + hb[w * CO + c];
        ls += hv; lq += hv * hv;
        hraw[(size_t)w * CO + c] = hv;
        const int w2 = 16 + r + 8 * hi;
        if (w2 < VV) {
          const float hv2 = hc1[r] + hb[w2 * CO + c];
          ls += hv2; lq += hv2 * hv2;
          hraw[(size_t)w2 * CO + c] = hv2;
        }
      }
      ls += __shfl_xor(ls, 16);
      lq += __shfl_xor(lq, 16);
      if (hi == 0) {
        atomicAdd(&s_stats[c], ls);
        atomicAdd(&s_stats[128 + c], lq);
      }
    }
    // residual GEMM (stats only; recomputed in K5)
    {
      v8f rc0 = {}; v8f rc1 = {};
      for (int ks = 0; ks < 16; ++ks) {
        const int i0 = ks * 4 + 2 * hi;
        v2f a0 = *(const v2f*)&s_x[wave][l16 * CIN + i0];
        v2f a1 = *(const v2f*)&s_x[wave][(l16 + 16) * CIN + i0];
        const float* bp = wrt + (size_t)i0 * CO + c;
        v2f bb; bb.x = bp[0]; bb.y = bp[CO];
        rc0 = wmma4(a0, bb, rc0);
        rc1 = wmma4(a1, bb, rc1);
      }
      const float br = b_res[c];
      float ls = 0.f, lq = 0.f;
#pragma unroll
      for (int r = 0; r < 8; ++r) {
        const float rv = rc0[r] + br;
        ls += rv; lq += rv * rv;
        const int w2 = 16 + r + 8 * hi;
        if (w2 < VV) {
          const float rv2 = rc1[r] + br;
          ls += rv2; lq += rv2 * rv2;
        }
      }
      ls += __shfl_xor(ls, 16);
      lq += __shfl_xor(lq, 16);
      if (hi == 0) {
        atomicAdd(&s_stats[256 + c], ls);
        atomicAdd(&s_stats[384 + c], lq);
      }
    }
  }
  __syncthreads();
  if (tid < 128) {
    atomicAdd(&ws[SUM_H + tid], s_stats[tid]);
    atomicAdd(&ws[SQ_H  + tid], s_stats[128 + tid]);
    atomicAdd(&ws[SUM_R + tid], s_stats[256 + tid]);
    atomicAdd(&ws[SQ_R  + tid], s_stats[384 + tid]);
  }
}

// ===================== K2 / K4: finalize BN params =====================
__global__ void k2_fin12(const float* __restrict__ gamma, const float* __restrict__ beta,
                         float* __restrict__ ws) {
  const int c = threadIdx.x;
  if (c >= 128) return;
  const float n = (float)NROWS;
  float m = ws[SUM_H + c] / n;
  float v = ws[SQ_H + c] / n - m * m;
  float s = gamma[c] * rsqrtf(v + EPSV);
  ws[SC2 + c] = s; ws[SH2 + c] = beta[c] - m * s;
  m = ws[SUM_R + c] / n;
  v = ws[SQ_R + c] / n - m * m;
  s = gamma[c] * rsqrtf(v + EPSV);
  ws[SC1 + c] = s; ws[SH1 + c] = beta[c] - m * s;
}

__global__ void k4_fin3(const float* __restrict__ gamma, const float* __restrict__ beta,
                        float* __restrict__ ws) {
  const int c = threadIdx.x;
  if (c >= 128) return;
  const float n = (float)NROWS;
  const float m = ws[SUM_T + c] / n;
  const float v = ws[SQ_T + c] / n - m * m;
  const float s = gamma[c] * rsqrtf(v + EPSV);
  ws[SC3 + c] = s; ws[SH3 + c] = beta[c] - m * s;
}

// ================= K3: TCN as implicit GEMM [64x1152]x[1152x128] =================
// grid = B*V*(T/64) = 6400, block = 256 (8 waves = 4 t-subtiles x 2 channel halves)
// Weight chunks double-buffered into LDS via the Tensor Data Mover (TENSORcnt).
#define WCH 16                      // weight chunk rows (K-slice)
#define NCHK (KER * CO / WCH)       // 72 chunks

__global__ void __launch_bounds__(256) k3_tcn(const float* __restrict__ b_tcn,
                                              float* __restrict__ ws,
                                              float* __restrict__ out) {
  __shared__ float s_hn[72 * CO];      // 36KB: relu(bn2(h_raw)) halo tile
  __shared__ float s_w[2][WCH * CO];   // 2 x 8KB: double-buffered weight chunks
  __shared__ float s_bn[2 * 128];      // scale2/shift2
  __shared__ float s_stats[2 * 128];

  const int tid  = threadIdx.x;
  const int wave = tid >> 5;
  const int lane = tid & 31;
  const int l16  = lane & 15;
  const int hi   = lane >> 4;
  const int tsub = wave & 3;
  const int nhalf = wave >> 2;

  const int bid = blockIdx.x;
  const int b   = bid / (VV * 16);
  const int rem = bid % (VV * 16);
  const int v   = rem >> 4;
  const int t0  = (rem & 15) * 64;

  for (int i = tid; i < 256; i += 256) {
    s_bn[i] = (i < 128) ? ws[SC2 + i] : ws[SH2 + (i - 128)];
    s_stats[i] = 0.f;
  }
  __syncthreads();
  // stage hn tile: rows rt in [0,72) <-> global t = t0 - 4 + rt
  {
    const float* hraw = ws + HRAW_OFF;
    for (int e = tid; e < 72 * CO; e += 256) {
      const int rt = e >> 7, c = e & 127;
      const int tg = t0 - 4 + rt;
      float val = 0.f;
      if (tg >= 0 && tg < TT) {
        const float h = hraw[(((size_t)(b * TT + tg)) * VV + v) * CO + c];
        val = fmaxf(h * s_bn[c] + s_bn[128 + c], 0.f);
      }
      s_hn[e] = val;
    }
  }

  v8f acc0 = {}, acc1 = {}, acc2 = {}, acc3 = {};
  const float* wtt = ws + WTT_OFF;

#ifdef USE_TDM
  if (wave == 0) tdm_load_rows(wtt, &s_w[0][0], WCH);   // prefetch chunk 0
#endif

  for (int ch = 0; ch < NCHK; ++ch) {
    const int j0c = ch * WCH;
    const int cur = ch & 1;
#ifdef USE_TDM
    if (wave == 0) __builtin_amdgcn_s_wait_tensorcnt(0);  // chunk ch landed
    __syncthreads();                                      // visible to all waves
    if (wave == 0 && ch + 1 < NCHK)                       // overlap next DMA
      tdm_load_rows(wtt + (size_t)(j0c + WCH) * CO, &s_w[cur ^ 1][0], WCH);
#else
    __syncthreads();
    for (int e = tid; e < WCH * CO; e += 256)
      s_w[cur][e] = wtt[(size_t)j0c * CO + e];
    __syncthreads();
#endif
#pragma unroll
    for (int js = 0; js < WCH / 4; ++js) {
      const int j0 = j0c + js * 4;
      const int r  = j0 >> 7;                // kernel tap (const within k-step)
      const int c0 = j0 & 127;
      const int rt = tsub * 16 + l16 + r;
      v2f a = *(const v2f*)&s_hn[rt * CO + c0 + 2 * hi];
      const int jj = js * 4 + 2 * hi;
      {
        const int oc = nhalf * 64 + l16;
        v2f bb; bb.x = s_w[cur][jj * CO + oc]; bb.y = s_w[cur][(jj + 1) * CO + oc];
        acc0 = wmma4(a, bb, acc0);
      }
      {
        const int oc = nhalf * 64 + 16 + l16;
        v2f bb; bb.x = s_w[cur][jj * CO + oc]; bb.y = s_w[cur][(jj + 1) * CO + oc];
        acc1 = wmma4(a, bb, acc1);
      }
      {
        const int oc = nhalf * 64 + 32 + l16;
        v2f bb; bb.x = s_w[cur][jj * CO + oc]; bb.y = s_w[cur][(jj + 1) * CO + oc];
        acc2 = wmma4(a, bb, acc2);
      }
      {
        const int oc = nhalf * 64 + 48 + l16;
        v2f bb; bb.x = s_w[cur][jj * CO + oc]; bb.y = s_w[cur][(jj + 1) * CO + oc];
        acc3 = wmma4(a, bb, acc3);
      }
    }
  }

  // bias + stats + store raw TCN output to d_out
#pragma unroll
  for (int nt = 0; nt < 4; ++nt) {
    const v8f acc = (nt == 0) ? acc0 : (nt == 1) ? acc1 : (nt == 2) ? acc2 : acc3;
    const int oc = nhalf * 64 + nt * 16 + l16;
    const float bias = b_tcn[oc];
    float ls = 0.f, lq = 0.f;
#pragma unroll
    for (int r8 = 0; r8 < 8; ++r8) {
      const int tg = t0 + tsub * 16 + r8 + 8 * hi;
      const float val = acc[r8] + bias;
      ls += val; lq += val * val;
      out[(((size_t)(b * TT + tg)) * VV + v) * CO + oc] = val;
    }
    ls += __shfl_xor(ls, 16);
    lq += __shfl_xor(lq, 16);
    if (hi == 0) {
      atomicAdd(&s_stats[oc], ls);
      atomicAdd(&s_stats[128 + oc], lq);
    }
  }
  __syncthreads();
  if (tid < 128) {
    atomicAdd(&ws[SUM_T + tid], s_stats[tid]);
    atomicAdd(&ws[SQ_T + tid], s_stats[128 + tid]);
  }
}

// ========== K5: out = relu(bn3(out_raw) + bn1(x @ w_res^T + b_res)) ==========
// grid = NROWS/128 = 3200, block = 256 (8 waves x 16 rows)
__global__ void __launch_bounds__(256) k5_final(const float* __restrict__ x,
                                                const float* __restrict__ b_res,
                                                float* __restrict__ ws,
                                                float* __restrict__ out) {
  __shared__ float s_x[128 * CIN];   // 32KB
  const int tid  = threadIdx.x;
  const int wave = tid >> 5;
  const int lane = tid & 31;
  const int l16  = lane & 15;
  const int hi   = lane >> 4;
  const size_t row0 = (size_t)blockIdx.x * 128;

  for (int e = tid; e < 128 * CIN; e += 256) s_x[e] = x[row0 * CIN + e];
  __syncthreads();

  const float* wrt = ws + WRT_OFF;
  const int rl = wave * 16 + l16;

  for (int nt = 0; nt < 8; ++nt) {
    const int c = nt * 16 + l16;
    v8f acc = {};
    for (int ks = 0; ks < 16; ++ks) {
      const int i0 = ks * 4 + 2 * hi;
      v2f a = *(const v2f*)&s_x[rl * CIN + i0];
      const float* bp = wrt + (size_t)i0 * CO + c;
      v2f bb; bb.x = bp[0]; bb.y = bp[CO];
      acc = wmma4(a, bb, acc);
    }
    const float s1 = ws[SC1 + c], t1 = ws[SH1 + c];
    const float s3 = ws[SC3 + c], t3 = ws[SH3 + c];
    const float br = b_res[c];
#pragma unroll
    for (int r8 = 0; r8 < 8; ++r8) {
      const size_t gr = row0 + (size_t)wave * 16 + r8 + 8 * hi;
      const size_t idx = gr * CO + c;
      const float tv = out[idx];                  // raw TCN output
      const float rv = acc[r8] + br;              // raw residual
      const float o = s3 * tv + t3 + s1 * rv + t1;
      out[idx] = fmaxf(o, 0.f);
    }
  }
}

// ============================ launcher ============================
extern "C" void kernel_launch(void* const* d_in, const int* in_sizes, int n_in,
                              void* d_out, int out_size, void* d_ws, size_t ws_size,
                              hipStream_t stream) {
  const float* x      = (const float*)d_in[0];
  const float* adj    = (const float*)d_in[1];
  const float* edge   = (const float*)d_in[2];
  const float* w_sp   = (const float*)d_in[3];
  const float* b_sp   = (const float*)d_in[4];
  const float* w_tcn  = (const float*)d_in[5];
  const float* b_tcn  = (const float*)d_in[6];
  const float* w_res  = (const float*)d_in[7];
  const float* b_res  = (const float*)d_in[8];
  const float* gamma  = (const float*)d_in[9];
  const float* beta   = (const float*)d_in[10];
  float* out = (float*)d_out;
  float* ws  = (float*)d_ws;

  k0_prep<<<dim3(128), dim3(256), 0, stream>>>(adj, edge, w_sp, b_sp, w_tcn, w_res, ws);
  k1_spatial<<<dim3(BB * TT / 4), dim3(128), 0, stream>>>(x, adj, edge, b_res, ws);
  k2_fin12<<<dim3(1), dim3(128), 0, stream>>>(gamma, beta, ws);
  k3_tcn<<<dim3(BB * VV * (TT / 64)), dim3(256), 0, stream>>>(b_tcn, ws, out);
  k4_fin3<<<dim3(1), dim3(128), 0, stream>>>(gamma, beta, ws);
  k5_final<<<dim3(NROWS / 128), dim3(256), 0, stream>>>(x, b_res, ws, out);
}